// GCN_22041772163466
// MI455X (gfx1250) — compile-verified
//
#include <hip/hip_runtime.h>
#include <hip/hip_bf16.h>

typedef __attribute__((ext_vector_type(2))) float v2f;
typedef __attribute__((ext_vector_type(8))) float v8f;

// ---------------------------------------------------------------------------
// Degree / normalization kernels
// ---------------------------------------------------------------------------
__global__ void init_deg_kernel(float* __restrict__ deg, int n) {
    int i = blockIdx.x * blockDim.x + threadIdx.x;
    if (i < n) deg[i] = 1.0f;   // self-loop contributes 1 to every node
}

__global__ void deg_accum_kernel(const long long* __restrict__ dst,
                                 float* __restrict__ deg, long long E) {
    long long e = (long long)blockIdx.x * blockDim.x + threadIdx.x;
    if (e < E) atomicAdd(&deg[dst[e]], 1.0f);
}

__global__ void make_dinv_kernel(float* __restrict__ deg, int n) {
    int i = blockIdx.x * blockDim.x + threadIdx.x;
    if (i < n) deg[i] = rsqrtf(deg[i]);   // deg >= 1 always (self loop)
}

// ---------------------------------------------------------------------------
// WMMA f32 GEMM:  C[M,N] = A[M,256] * B[256,N]
// One block = 16 rows of C, 256 threads = 8 waves.
// Wave w covers columns [w*16*TPW, (w+1)*16*TPW); TPW = n-tiles per wave.
// Requires M % 16 == 0 (M = 100000 = 6250*16).
// ---------------------------------------------------------------------------
template <int N, int TPW>
__global__ void gemm_wmma_f32_kernel(const float* __restrict__ A,
                                     const float* __restrict__ B,
                                     float* __restrict__ C, int M) {
    constexpr int K = 256;
    const int wave = threadIdx.x >> 5;
    const int lane = threadIdx.x & 31;
    const int laneLo = lane & 15;          // M (for A) / N (for B,C) index
    const int kHalf  = (lane >> 4) << 1;   // 0 for lanes 0-15, 2 for lanes 16-31
    const int mBase  = blockIdx.x << 4;
    const long long aRow = (long long)(mBase + laneLo) * K;

    v8f acc[TPW] = {};

    for (int k = 0; k < K; k += 4) {
        // A fragment: VGPR0 = K=k+kHalf, VGPR1 = K=k+kHalf+1 (ISA 16x4 f32 layout)
        v2f a = *(const v2f*)(A + aRow + k + kHalf);
#pragma unroll
        for (int t = 0; t < TPW; ++t) {
            const int col = (wave * TPW + t) * 16 + laneLo;
            v2f b;
            b.x = B[(long long)(k + kHalf) * N + col];
            b.y = B[(long long)(k + kHalf + 1) * N + col];
            acc[t] = __builtin_amdgcn_wmma_f32_16x16x4_f32(
                false, a, false, b, (short)0, acc[t], false, false);
        }
    }

    // C/D layout: VGPR r -> row (r + 0/8 by lane half), col = laneLo
    const int mOff = (lane >> 4) << 3;  // 0 or 8
#pragma unroll
    for (int t = 0; t < TPW; ++t) {
        const int col = (wave * TPW + t) * 16 + laneLo;
#pragma unroll
        for (int r = 0; r < 8; ++r) {
            C[(long long)(mBase + mOff + r) * N + col] = acc[t][r];
        }
    }
}

// ---------------------------------------------------------------------------
// Edge scatter:  AGG[dst] += H[src] * dinv[src]*dinv[dst]
// D/4 threads per edge, float4 gather, 4 f32 atomics per thread.
// ---------------------------------------------------------------------------
template <int D>
__global__ void scatter_edges_kernel(const long long* __restrict__ src,
                                     const long long* __restrict__ dst,
                                     const float* __restrict__ dinv,
                                     const float* __restrict__ H,
                                     float* __restrict__ AGG, long long E) {
    constexpr int TPE = D / 4;  // threads per edge (power of two)
    long long gid = (long long)blockIdx.x * blockDim.x + threadIdx.x;
    long long e = gid / TPE;
    if (e >= E) return;
    int j = (int)(gid & (TPE - 1)) * 4;
    long long s = src[e];
    long long d = dst[e];
    float coef = dinv[s] * dinv[d];
    float4 v = *(const float4*)(H + s * (long long)D + j);
    float* o = AGG + d * (long long)D + j;
    atomicAdd(o + 0, v.x * coef);
    atomicAdd(o + 1, v.y * coef);
    atomicAdd(o + 2, v.z * coef);
    atomicAdd(o + 3, v.w * coef);
}

// ---------------------------------------------------------------------------
// Epilogues: add self-loop term (dinv^2 * H) + bias, then activation.
// ---------------------------------------------------------------------------
template <int D>
__global__ void post_relu_kernel(const float* __restrict__ H,
                                 const float* __restrict__ dinv,
                                 const float* __restrict__ bias,
                                 float* __restrict__ AGG, long long total) {
    long long idx = (long long)blockIdx.x * blockDim.x + threadIdx.x;
    if (idx >= total) return;
    long long row = idx / D;
    int col = (int)(idx & (D - 1));
    float di = dinv[row];
    float v = AGG[idx] + di * di * H[idx] + bias[col];
    AGG[idx] = v > 0.0f ? v : 0.0f;
}

template <int D>
__global__ void post_sigmoid_kernel(const float* __restrict__ H,
                                    const float* __restrict__ dinv,
                                    const float* __restrict__ bias,
                                    const float* __restrict__ AGG,
                                    float* __restrict__ out, long long total) {
    long long idx = (long long)blockIdx.x * blockDim.x + threadIdx.x;
    if (idx >= total) return;
    long long row = idx / D;
    int col = (int)(idx & (D - 1));
    float di = dinv[row];
    float v = AGG[idx] + di * di * H[idx] + bias[col];
    out[idx] = 1.0f / (1.0f + expf(-v));
}

// ---------------------------------------------------------------------------
// Launch
// ---------------------------------------------------------------------------
extern "C" void kernel_launch(void* const* d_in, const int* in_sizes, int n_in,
                              void* d_out, int out_size, void* d_ws, size_t ws_size,
                              hipStream_t stream) {
    const float* x        = (const float*)d_in[0];
    const long long* ei   = (const long long*)d_in[1];  // int64 [2, E] row-major
    const float* W1       = (const float*)d_in[2];
    const float* b1       = (const float*)d_in[3];
    const float* W2       = (const float*)d_in[4];
    const float* b2       = (const float*)d_in[5];
    float* out            = (float*)d_out;

    const int M = in_sizes[0] / 256;          // 100000 nodes (multiple of 16)
    const long long E = in_sizes[1] / 2;      // 3.2M edges
    const long long* srcp = ei;
    const long long* dstp = ei + E;

    // Workspace layout (floats)
    float* ws = (float*)d_ws;
    size_t o = 0;
    float* dinv = ws + o; o += (((size_t)M + 127) & ~(size_t)127);
    float* H1   = ws + o; o += (size_t)M * 256;
    float* A1   = ws + o; o += (size_t)M * 256;
    float* H2   = ws + o; o += (size_t)M * 128;
    float* A2   = ws + o; o += (size_t)M * 128;

    hipMemsetAsync(A1, 0, (size_t)M * 256 * sizeof(float), stream);
    hipMemsetAsync(A2, 0, (size_t)M * 128 * sizeof(float), stream);

    // Normalization: deg -> dinv
    init_deg_kernel<<<(M + 255) / 256, 256, 0, stream>>>(dinv, M);
    deg_accum_kernel<<<(unsigned)((E + 255) / 256), 256, 0, stream>>>(dstp, dinv, E);
    make_dinv_kernel<<<(M + 255) / 256, 256, 0, stream>>>(dinv, M);

    // Layer 1: H1 = X @ W1 (N=256, 2 n-tiles/wave)
    gemm_wmma_f32_kernel<256, 2><<<M / 16, 256, 0, stream>>>(x, W1, H1, M);
    {
        long long tot = E * (256 / 4);
        scatter_edges_kernel<256><<<(unsigned)((tot + 255) / 256), 256, 0, stream>>>(
            srcp, dstp, dinv, H1, A1, E);
    }
    {
        long long tot = (long long)M * 256;
        post_relu_kernel<256><<<(unsigned)((tot + 255) / 256), 256, 0, stream>>>(
            H1, dinv, b1, A1, tot);
    }

    // Layer 2: H2 = A1 @ W2 (N=128, 1 n-tile/wave)
    gemm_wmma_f32_kernel<128, 1><<<M / 16, 256, 0, stream>>>(A1, W2, H2, M);
    {
        long long tot = E * (128 / 4);
        scatter_edges_kernel<128><<<(unsigned)((tot + 255) / 256), 256, 0, stream>>>(
            srcp, dstp, dinv, H2, A2, E);
    }
    {
        long long tot = (long long)M * 128;
        post_sigmoid_kernel<128><<<(unsigned)((tot + 255) / 256), 256, 0, stream>>>(
            H2, dinv, b2, A2, out, tot);
    }
}